// CircleLoss_2525440770464
// MI455X (gfx1250) — compile-verified
//
#include <hip/hip_runtime.h>
#include <math.h>

// CircleLoss fused pipeline for MI455X (gfx1250, wave32, WMMA)
//   K1: row L2-normalize (4096 x 512 f32)
//   K2: lower-triangular 64x64 sim tiles via V_WMMA_F32_16X16X4_F32 from a
//       full-depth 258KB LDS stage loaded with GLOBAL_LOAD_ASYNC_TO_LDS_B128
//       (ASYNCcnt path); fused pos/neg term eval + online LSE, off-diagonal
//       blocks weighted x2 (symmetry)
//   K3: merge 2080 per-block (m,s) pairs -> softplus(lse_pos + lse_neg)

#define N_ROWS 4096
#define DIMS   512
#define TILE   64
#define LDS_STRIDE 516                      // 512 + 4 pad: conflict-free fragment reads
#define NTILE  (N_ROWS / TILE)              // 64
#define NBLK_TRI (NTILE * (NTILE + 1) / 2)  // 2080
#define MARGIN 0.25f
#define GAMMA  256.0f

typedef __attribute__((ext_vector_type(2))) float v2f;
typedef __attribute__((ext_vector_type(8))) float v8f;

__device__ __forceinline__ void lse_add(float& m, float& s, float x, float w) {
    // online logsumexp accumulate: s = sum w*exp(x - m)
    if (x > m) {
        s = s * expf(m - x) + w;   // first hit: s==0, expf(-inf)==0 -> s=w
        m = x;
    } else {
        s += w * expf(x - m);
    }
}

__device__ __forceinline__ void lse_merge(float& m, float& s, float m2, float s2) {
    if (s2 == 0.0f) return;
    if (s == 0.0f) { m = m2; s = s2; return; }
    if (m2 > m) { s = s * expf(m - m2) + s2; m = m2; }
    else        { s += s2 * expf(m2 - m); }
}

// ---------------- Kernel 1: L2 row normalize ----------------
__global__ __launch_bounds__(256) void normalize_rows(const float* __restrict__ f,
                                                      float* __restrict__ fn) {
    __shared__ float red[256];
    const int row = blockIdx.x;
    const float* src = f + (size_t)row * DIMS;
    float ss = 0.0f;
    for (int c = threadIdx.x; c < DIMS; c += 256) {
        float v = src[c];
        ss += v * v;
    }
    red[threadIdx.x] = ss;
    __syncthreads();
    for (int off = 128; off > 0; off >>= 1) {
        if (threadIdx.x < off) red[threadIdx.x] += red[threadIdx.x + off];
        __syncthreads();
    }
    const float inv = 1.0f / fmaxf(sqrtf(red[0]), 1e-12f);
    float* dst = fn + (size_t)row * DIMS;
    for (int c = threadIdx.x; c < DIMS; c += 256) dst[c] = src[c] * inv;
}

// ---------------- Kernel 2: fused sim-tile + online LSE ----------------
__global__ __launch_bounds__(256) void circle_tile(const float* __restrict__ fn,
                                                   const int* __restrict__ labels,
                                                   float* __restrict__ partials) {
    extern __shared__ float smem[];
    float* ldsA = smem;                                  // [TILE][LDS_STRIDE]
    float* ldsB = smem + TILE * LDS_STRIDE;              // [TILE][LDS_STRIDE]
    int*   lrow = (int*)(smem + 2 * TILE * LDS_STRIDE);  // [TILE]
    int*   lcol = lrow + TILE;                           // [TILE]

    // decode linear block id -> lower-triangular (bi >= bj) tile coords
    const int p = blockIdx.x;
    int bi = (int)((sqrtf(8.0f * (float)p + 1.0f) - 1.0f) * 0.5f);
    while ((bi + 1) * (bi + 2) / 2 <= p) ++bi;
    while (bi * (bi + 1) / 2 > p) --bi;
    const int bj = p - bi * (bi + 1) / 2;
    const int rowBase = bi * TILE;
    const int colBase = bj * TILE;
    const float wgt = (bi == bj) ? 1.0f : 2.0f;  // symmetry: count (i,j) and (j,i)

    const int tid = threadIdx.x;

    // cooperative full-depth stage via async global->LDS (ASYNCcnt, no VGPR
    // round-trip). Dynamic LDS starts at byte offset 0, so raw LDS byte
    // offsets address the same storage the ds_load fragment reads use.
    {
        const float* gA = fn + (size_t)rowBase * DIMS;
        const float* gB = fn + (size_t)colBase * DIMS;
        for (int idx = tid; idx < TILE * (DIMS / 4); idx += 256) {
            const int r  = idx >> 7;        // / (DIMS/4)
            const int c4 = idx & 127;       // % (DIMS/4)
            unsigned ldsOffA = (unsigned)((r * LDS_STRIDE + c4 * 4) * 4);
            unsigned ldsOffB = ldsOffA + (unsigned)(TILE * LDS_STRIDE * 4);
            unsigned long long ga =
                (unsigned long long)(gA + (size_t)r * DIMS + c4 * 4);
            unsigned long long gb =
                (unsigned long long)(gB + (size_t)r * DIMS + c4 * 4);
            asm volatile("global_load_async_to_lds_b128 %0, %1, off"
                         :: "v"(ldsOffA), "v"(ga) : "memory");
            asm volatile("global_load_async_to_lds_b128 %0, %1, off"
                         :: "v"(ldsOffB), "v"(gb) : "memory");
        }
#if __has_builtin(__builtin_amdgcn_s_wait_asynccnt)
        __builtin_amdgcn_s_wait_asynccnt(0);
#else
        asm volatile("s_wait_asynccnt 0x0" ::: "memory");
#endif
    }
    if (tid < TILE)          lrow[tid]        = labels[rowBase + tid];
    else if (tid < 2 * TILE) lcol[tid - TILE] = labels[colBase + tid - TILE];
    __syncthreads();

    // wave -> two adjacent 16x16 subtiles (A-fragment reuse)
    const int wv   = tid >> 5;
    const int lane = tid & 31;
    const int m16  = lane & 15;
    const int kh   = lane >> 4;          // K half per ISA 16x16x4 f32 layout
    const int tr   = wv >> 1;            // subtile row 0..3
    const int tc0  = (wv & 1) * 2;       // subtile cols tc0, tc0+1

    const float* arow  = ldsA + (tr  * 16 + m16) * LDS_STRIDE + 2 * kh;
    const float* brow0 = ldsB + (tc0 * 16 + m16) * LDS_STRIDE + 2 * kh;
    const float* brow1 = brow0 + 16 * LDS_STRIDE;

    v8f acc0 = {};
    v8f acc1 = {};
#pragma unroll 8
    for (int k0 = 0; k0 < DIMS; k0 += 4) {
        v2f a  = *(const v2f*)(arow  + k0);
        v2f b0 = *(const v2f*)(brow0 + k0);
        v2f b1 = *(const v2f*)(brow1 + k0);
        acc0 = __builtin_amdgcn_wmma_f32_16x16x4_f32(false, a, false, b0,
                                                     (short)0, acc0, false, false);
        acc1 = __builtin_amdgcn_wmma_f32_16x16x4_f32(false, a, false, b1,
                                                     (short)0, acc1, false, false);
    }

    // fused CircleLoss terms + per-thread online LSE over fragment elements
    float mp = -INFINITY, sp = 0.0f, mn = -INFINITY, sn = 0.0f;
    const int lj0 = lcol[tc0 * 16 + m16];
    const int lj1 = lcol[(tc0 + 1) * 16 + m16];
#pragma unroll
    for (int r = 0; r < 8; ++r) {
        const int li = lrow[tr * 16 + 8 * kh + r];
        const float s0 = acc0[r];
        const float s1 = acc1[r];
        if (li == lj0) lse_add(mp, sp, -fmaxf(1.0f + MARGIN - s0, 0.0f) * s0 * GAMMA, wgt);
        else           lse_add(mn, sn,  fmaxf(s0 + MARGIN, 0.0f) * s0 * GAMMA, wgt);
        if (li == lj1) lse_add(mp, sp, -fmaxf(1.0f + MARGIN - s1, 0.0f) * s1 * GAMMA, wgt);
        else           lse_add(mn, sn,  fmaxf(s1 + MARGIN, 0.0f) * s1 * GAMMA, wgt);
    }

    // block reduction (reuse LDS stage space)
    __syncthreads();
    float4* redbuf = (float4*)smem;
    redbuf[tid] = make_float4(mp, sp, mn, sn);
    __syncthreads();
    for (int off = 128; off > 0; off >>= 1) {
        if (tid < off) {
            float4 a = redbuf[tid];
            float4 o = redbuf[tid + off];
            lse_merge(a.x, a.y, o.x, o.y);
            lse_merge(a.z, a.w, o.z, o.w);
            redbuf[tid] = a;
        }
        __syncthreads();
    }
    if (tid == 0) {
        float4 a = redbuf[0];
        partials[4 * blockIdx.x + 0] = a.x;
        partials[4 * blockIdx.x + 1] = a.y;
        partials[4 * blockIdx.x + 2] = a.z;
        partials[4 * blockIdx.x + 3] = a.w;
    }
}

// ---------------- Kernel 3: final merge + softplus ----------------
__global__ __launch_bounds__(256) void finalize(const float* __restrict__ partials,
                                                float* __restrict__ out) {
    __shared__ float4 red[256];
    float mp = -INFINITY, sp = 0.0f, mn = -INFINITY, sn = 0.0f;
    for (int i = threadIdx.x; i < NBLK_TRI; i += 256) {
        lse_merge(mp, sp, partials[4 * i + 0], partials[4 * i + 1]);
        lse_merge(mn, sn, partials[4 * i + 2], partials[4 * i + 3]);
    }
    red[threadIdx.x] = make_float4(mp, sp, mn, sn);
    __syncthreads();
    for (int off = 128; off > 0; off >>= 1) {
        if (threadIdx.x < off) {
            float4 a = red[threadIdx.x];
            float4 o = red[threadIdx.x + off];
            lse_merge(a.x, a.y, o.x, o.y);
            lse_merge(a.z, a.w, o.z, o.w);
            red[threadIdx.x] = a;
        }
        __syncthreads();
    }
    if (threadIdx.x == 0) {
        float4 a = red[0];
        const float lse_pos = (a.y > 0.0f) ? a.x + logf(a.y) : -INFINITY;
        const float lse_neg = (a.w > 0.0f) ? a.z + logf(a.w) : -INFINITY;
        const float z = lse_pos + lse_neg;
        // stable softplus; z==-inf -> 0 naturally
        out[0] = (z > 0.0f) ? z + log1pf(expf(-z)) : log1pf(expf(z));
    }
}

extern "C" void kernel_launch(void* const* d_in, const int* in_sizes, int n_in,
                              void* d_out, int out_size, void* d_ws, size_t ws_size,
                              hipStream_t stream) {
    (void)in_sizes; (void)n_in; (void)out_size; (void)ws_size;
    const float* feats  = (const float*)d_in[0];
    const int*   labels = (const int*)d_in[1];
    float*       out    = (float*)d_out;

    float* fn       = (float*)d_ws;                       // 4096*512 f32 = 8 MB
    float* partials = fn + (size_t)N_ROWS * DIMS;         // 2080 * 4 f32

    normalize_rows<<<N_ROWS, 256, 0, stream>>>(feats, fn);

    const size_t smem_bytes = (size_t)(2 * TILE * LDS_STRIDE + 2 * TILE) * sizeof(float);
    circle_tile<<<NBLK_TRI, 256, smem_bytes, stream>>>(fn, labels, partials);

    finalize<<<1, 256, 0, stream>>>(partials, out);
}